// TemporalConsistencyLoss_78993038508697
// MI455X (gfx1250) — compile-verified
//
#include <hip/hip_runtime.h>

#define WEIGHT 0.1f
#define THRESH 0.3f
#define MARGIN 0.1f

typedef __attribute__((ext_vector_type(2))) float v2f;
typedef __attribute__((ext_vector_type(4))) float v4f;
typedef __attribute__((ext_vector_type(8))) float v8f;

// Full 32-lane sum using V_WMMA_F32_16X16X4_F32.
// A (16x4 f32, 2 VGPRs): VGPR0 = K0 (lanes 0-15) / K2 (lanes 16-31) -> value,
//                        VGPR1 = K1 / K3 -> zero.
// B = all ones (immune to B K-layout details).
// D[m][n] = s[m] + s[m+16]; summing the 8 D VGPRs leaves sum(rows 0..7) in
// lanes 0-15 and sum(rows 8..15) in lanes 16-31; one xor-16 swap finishes.
__device__ __forceinline__ float wave_sum_wmma(float v) {
    v2f a; a[0] = v;    a[1] = 0.0f;
    v2f b; b[0] = 1.0f; b[1] = 1.0f;
    v8f c = {};
    v8f d = __builtin_amdgcn_wmma_f32_16x16x4_f32(
        /*neg_a=*/false, a, /*neg_b=*/false, b,
        /*c_mod=*/(short)0, c, /*reuse_a=*/false, /*reuse_b=*/false);
    float s = ((d[0] + d[1]) + (d[2] + d[3])) + ((d[4] + d[5]) + (d[6] + d[7]));
    // ds_swizzle_b32 group-of-32: xor_mask=0x10, and_mask=0x1f -> SWAPX16.
    s += __int_as_float(__builtin_amdgcn_ds_swizzle(__float_as_int(s), 0x401F));
    return s;  // full sum in every lane
}

__device__ __forceinline__ void tcl_accum(const v4f* __restrict__ pred,
                                          const float* __restrict__ times,
                                          int i, float& total, float& cnt) {
    // Single-pass stream: non-temporal, keep it out of L2.
    v4f   p = __builtin_nontemporal_load(&pred[i]);
    float t = __builtin_nontemporal_load(&times[i]);

    float p4 = p[1], p5 = p[2], p6 = p[3];

    bool g4 = p4 > THRESH, g5 = p5 > THRESH, g6 = p6 > THRESH;
    bool m45  = g4 && g5;
    bool m56  = g5 && g6;
    bool m456 = m45 && g6;

    float term45 = fmaxf(MARGIN - (p5 * t - p4 * t), 0.0f);
    float term56 = fmaxf(MARGIN - (p6 * t - p5 * t), 0.0f);
    float d45 = fabsf(p5 - p4);
    float d56 = fabsf(p6 - p5);
    float term_o = fmaxf(d45 - d56 + MARGIN, 0.0f);

    total += (m45  ? term45 : 0.0f)
           + (m56  ? term56 : 0.0f)
           + (m456 ? term_o : 0.0f);
    cnt   += (m45  ? 1.0f : 0.0f)
           + (m56  ? 1.0f : 0.0f)
           + (m456 ? 1.0f : 0.0f);
}

__global__ void tcl_init(float* ws) {
    if (threadIdx.x < 2) ws[threadIdx.x] = 0.0f;
}

__global__ __launch_bounds__(256) void tcl_main(const v4f* __restrict__ pred,
                                                const float* __restrict__ times,
                                                float* __restrict__ ws, int n) {
    float total = 0.0f;
    float cnt   = 0.0f;

    const int tid      = blockIdx.x * blockDim.x + threadIdx.x;
    const int nthreads = gridDim.x * blockDim.x;
    const int sweeps   = n / nthreads;   // uniform, branch-free main loop
    const int rem      = n - sweeps * nthreads;

    int i = tid;
    #pragma unroll 4
    for (int k = 0; k < sweeps; ++k, i += nthreads)
        tcl_accum(pred, times, i, total, cnt);
    if (tid < rem)
        tcl_accum(pred, times, i, total, cnt);

    // All threads reach here unconditionally -> EXEC all ones for WMMA.
    float wt = wave_sum_wmma(total);
    float wc = wave_sum_wmma(cnt);

    __shared__ float s_tot[8];
    __shared__ float s_cnt[8];
    const int wave = threadIdx.x >> 5;
    const int lane = threadIdx.x & 31;
    if (lane == 0) { s_tot[wave] = wt; s_cnt[wave] = wc; }
    __syncthreads();
    if (threadIdx.x == 0) {
        float bt = 0.0f, bc = 0.0f;
        #pragma unroll
        for (int w = 0; w < 8; ++w) { bt += s_tot[w]; bc += s_cnt[w]; }
        atomicAdd(&ws[0], bt);
        atomicAdd(&ws[1], bc);
    }
}

__global__ void tcl_final(const float* __restrict__ ws, float* __restrict__ out) {
    float total = ws[0];
    float count = ws[1];
    float loss = (count > 0.0f) ? (total / fmaxf(count, 1.0f)) : total;
    out[0] = WEIGHT * loss;
}

extern "C" void kernel_launch(void* const* d_in, const int* in_sizes, int n_in,
                              void* d_out, int out_size, void* d_ws, size_t ws_size,
                              hipStream_t stream) {
    const v4f*   pred  = (const v4f*)d_in[0];     // predictions: (B,4) f32
    const float* times = (const float*)d_in[1];   // relative_times: (B,1) f32
    float*       ws    = (float*)d_ws;            // [0]=total, [1]=count
    float*       out   = (float*)d_out;           // scalar f32

    const int n = in_sizes[1];                    // B rows

    tcl_init<<<1, 32, 0, stream>>>(ws);

    int blocks = (n + 256 * 8 - 1) / (256 * 8);   // ~8 rows per thread, one per sweep
    if (blocks > 4096) blocks = 4096;
    if (blocks < 1)    blocks = 1;
    tcl_main<<<blocks, 256, 0, stream>>>(pred, times, ws, n);

    tcl_final<<<1, 1, 0, stream>>>(ws, out);
}